// ThresholdDetector_62053687492841
// MI455X (gfx1250) — compile-verified
//
#include <hip/hip_runtime.h>
#include <hip/hip_bf16.h>
#include <stdint.h>

#define MAX_SPIKE 36
#define K_MAX 65536
#define TILE 8192
#define NBLOCKS 512
#define NTHREADS 256

typedef __attribute__((ext_vector_type(4))) unsigned int u32x4;
typedef __attribute__((ext_vector_type(8))) int          i32x8;
typedef __attribute__((ext_vector_type(4))) int          i32x4;

// ---------------------------------------------------------------------------
// Tensor Data Mover: async DMA of `nelem` f32 from global to LDS.
// D# built per CDNA5 ISA (group0 128b, group1 256b,
// groups 2/3 zero => <=2D tensor). EXEC is ignored by TDM; we issue from one
// lane and fence the workgroup with s_wait_tensorcnt + barrier.
// ---------------------------------------------------------------------------
__device__ __forceinline__ void tdm_load_f32(void* lds_dst, const float* gsrc,
                                             unsigned nelem) {
  unsigned long long ga = (unsigned long long)(uintptr_t)gsrc;
  unsigned lds = (unsigned)(uintptr_t)lds_dst;  // low 32 bits = LDS byte offset
  u32x4 g0;
  g0[0] = 1u;                                   // count=1, user descriptor
  g0[1] = lds;                                  // lds_addr
  g0[2] = (unsigned)(ga & 0xFFFFFFFFull);       // global_addr[31:0]
  g0[3] = (unsigned)((ga >> 32) & 0x01FFFFFFull) | (2u << 30);  // addr[56:32]|type=2
  i32x8 g1;
  g1[0] = (int)(2u << 16);                      // workgroup_mask=0, data_size=4B
  g1[1] = (int)((nelem & 0xFFFFu) << 16);       // tensor_dim0[15:0]
  g1[2] = (int)((nelem >> 16) | (1u << 16));    // tensor_dim0[31:16], tensor_dim1=1
  g1[3] = (int)((nelem & 0xFFFFu) << 16);       // tile_dim0
  g1[4] = 1;                                    // tile_dim1=1, tile_dim2=0
  g1[5] = (int)nelem;                           // tensor_dim0_stride[31:0]
  g1[6] = 0;
  g1[7] = 0;
  i32x4 z4; z4[0] = 0; z4[1] = 0; z4[2] = 0; z4[3] = 0;
#if __has_include(<hip/amd_detail/amd_gfx1250_TDM.h>)
  // amdgpu-toolchain (clang-23 / therock-10): 6-arg form
  i32x8 z8; z8[0]=0;z8[1]=0;z8[2]=0;z8[3]=0;z8[4]=0;z8[5]=0;z8[6]=0;z8[7]=0;
  __builtin_amdgcn_tensor_load_to_lds(g0, g1, z4, z4, z8, 0);
#else
  // ROCm 7.2 (clang-22): 5-arg form
  __builtin_amdgcn_tensor_load_to_lds(g0, g1, z4, z4, 0);
#endif
}

__device__ __forceinline__ int iminll(long long a, long long b) {
  return (int)(a < b ? a : b);
}

// ---------------------------------------------------------------------------
// Pass 1: 256-bin histogram of |x| exponent bits (bits[30:23]).
// Double-buffered TDM tiles; per-block LDS histogram; one global merge.
// ---------------------------------------------------------------------------
__global__ void hist_l1_kernel(const float* __restrict__ x, int n,
                               unsigned* __restrict__ hist) {
  __shared__ float buf[2][TILE];
  __shared__ unsigned lh[256];
  for (int i = threadIdx.x; i < 256; i += blockDim.x) lh[i] = 0u;

  const int tiles = (n + TILE - 1) / TILE;
  int t = blockIdx.x;
  int cur = 0;
  if (t < tiles && threadIdx.x == 0) {
    long long base = (long long)t * TILE;
    tdm_load_f32(buf[0], x + base, (unsigned)iminll(TILE, (long long)n - base));
  }
  __syncthreads();
  for (; t < tiles; t += gridDim.x) {
    long long base = (long long)t * TILE;
    int cnt = iminll(TILE, (long long)n - base);
    int nxt = t + gridDim.x;
    if (threadIdx.x == 0) {
      if (nxt < tiles) {
        long long nb = (long long)nxt * TILE;
        tdm_load_f32(buf[cur ^ 1], x + nb, (unsigned)iminll(TILE, (long long)n - nb));
        __builtin_amdgcn_s_wait_tensorcnt(1);   // oldest (current tile) done
      } else {
        __builtin_amdgcn_s_wait_tensorcnt(0);
      }
    }
    __syncthreads();
    for (int i = threadIdx.x; i < cnt; i += blockDim.x) {
      unsigned b = (__float_as_uint(buf[cur][i]) & 0x7FFFFFFFu) >> 23;
      atomicAdd(&lh[b], 1u);
    }
    __syncthreads();
    cur ^= 1;
  }
  __syncthreads();
  for (int i = threadIdx.x; i < 256; i += blockDim.x)
    if (lh[i]) atomicAdd(&hist[i], lh[i]);
}

__global__ void scan_l1_kernel(const unsigned* hist, unsigned* sel_bin,
                               unsigned* sel_rank, long long k0, long long k1) {
  long long ks[2] = {k0, k1};
  for (int j = 0; j < 2; ++j) {
    long long rem = ks[j];
    unsigned b = 0;
    while (b < 255u) {
      unsigned c = hist[b];
      if (rem < (long long)c) break;
      rem -= c; ++b;
    }
    sel_bin[j] = b;
    sel_rank[j] = (unsigned)rem;
  }
}

// ---------------------------------------------------------------------------
// Pass 2: for elements whose exponent bin matches either rank's bin,
// histogram mantissa bits[22:11] (4096 bins, two tracks, LDS).
// ---------------------------------------------------------------------------
__global__ void hist_l2_kernel(const float* __restrict__ x, int n,
                               const unsigned* __restrict__ sel_bin1,
                               unsigned* __restrict__ hist2) {
  __shared__ float buf[2][TILE];
  __shared__ unsigned lh[2 * 4096];
  const unsigned b1_0 = sel_bin1[0], b1_1 = sel_bin1[1];
  for (int i = threadIdx.x; i < 2 * 4096; i += blockDim.x) lh[i] = 0u;

  const int tiles = (n + TILE - 1) / TILE;
  int t = blockIdx.x;
  int cur = 0;
  if (t < tiles && threadIdx.x == 0) {
    long long base = (long long)t * TILE;
    tdm_load_f32(buf[0], x + base, (unsigned)iminll(TILE, (long long)n - base));
  }
  __syncthreads();
  for (; t < tiles; t += gridDim.x) {
    long long base = (long long)t * TILE;
    int cnt = iminll(TILE, (long long)n - base);
    int nxt = t + gridDim.x;
    if (threadIdx.x == 0) {
      if (nxt < tiles) {
        long long nb = (long long)nxt * TILE;
        tdm_load_f32(buf[cur ^ 1], x + nb, (unsigned)iminll(TILE, (long long)n - nb));
        __builtin_amdgcn_s_wait_tensorcnt(1);
      } else {
        __builtin_amdgcn_s_wait_tensorcnt(0);
      }
    }
    __syncthreads();
    for (int i = threadIdx.x; i < cnt; i += blockDim.x) {
      unsigned bits = __float_as_uint(buf[cur][i]) & 0x7FFFFFFFu;
      unsigned top8 = bits >> 23;
      unsigned mid = (bits >> 11) & 0xFFFu;
      if (top8 == b1_0) atomicAdd(&lh[mid], 1u);
      if (top8 == b1_1) atomicAdd(&lh[4096 + mid], 1u);
    }
    __syncthreads();
    cur ^= 1;
  }
  __syncthreads();
  for (int i = threadIdx.x; i < 2 * 4096; i += blockDim.x)
    if (lh[i]) atomicAdd(&hist2[i], lh[i]);
}

__global__ void scan_l2_kernel(const unsigned* hist2, const unsigned* sel_bin1,
                               const unsigned* sel_rank1, unsigned* sel_pre2,
                               unsigned* sel_rank2) {
  for (int j = 0; j < 2; ++j) {
    long long rem = (long long)sel_rank1[j];
    const unsigned* h = hist2 + j * 4096;
    unsigned b = 0;
    while (b < 4095u) {
      unsigned c = h[b];
      if (rem < (long long)c) break;
      rem -= c; ++b;
    }
    sel_pre2[j] = (sel_bin1[j] << 12) | b;
    sel_rank2[j] = (unsigned)rem;
  }
}

// ---------------------------------------------------------------------------
// Pass 3: elements matching top-20 prefix -> histogram bits[10:0].
// Matches are sparse (~1e3); direct global atomics.
// ---------------------------------------------------------------------------
__global__ void hist_l3_kernel(const float* __restrict__ x, int n,
                               const unsigned* __restrict__ sel_pre2,
                               unsigned* __restrict__ hist3) {
  __shared__ float buf[2][TILE];
  const unsigned p0 = sel_pre2[0], p1 = sel_pre2[1];

  const int tiles = (n + TILE - 1) / TILE;
  int t = blockIdx.x;
  int cur = 0;
  if (t < tiles && threadIdx.x == 0) {
    long long base = (long long)t * TILE;
    tdm_load_f32(buf[0], x + base, (unsigned)iminll(TILE, (long long)n - base));
  }
  __syncthreads();
  for (; t < tiles; t += gridDim.x) {
    long long base = (long long)t * TILE;
    int cnt = iminll(TILE, (long long)n - base);
    int nxt = t + gridDim.x;
    if (threadIdx.x == 0) {
      if (nxt < tiles) {
        long long nb = (long long)nxt * TILE;
        tdm_load_f32(buf[cur ^ 1], x + nb, (unsigned)iminll(TILE, (long long)n - nb));
        __builtin_amdgcn_s_wait_tensorcnt(1);
      } else {
        __builtin_amdgcn_s_wait_tensorcnt(0);
      }
    }
    __syncthreads();
    for (int i = threadIdx.x; i < cnt; i += blockDim.x) {
      unsigned bits = __float_as_uint(buf[cur][i]) & 0x7FFFFFFFu;
      unsigned pre = bits >> 11;
      unsigned lo = bits & 0x7FFu;
      if (pre == p0) atomicAdd(&hist3[lo], 1u);
      if (pre == p1) atomicAdd(&hist3[2048 + lo], 1u);
    }
    __syncthreads();
    cur ^= 1;
  }
}

__global__ void scan_l3_kernel(const unsigned* hist3, const unsigned* sel_pre2,
                               const unsigned* sel_rank2, float* thr_ws,
                               float* thr_out) {
  float v[2];
  for (int j = 0; j < 2; ++j) {
    long long rem = (long long)sel_rank2[j];
    const unsigned* h = hist3 + j * 2048;
    unsigned b = 0;
    while (b < 2047u) {
      unsigned c = h[b];
      if (rem < (long long)c) break;
      rem -= c; ++b;
    }
    v[j] = __uint_as_float((sel_pre2[j] << 11) | b);
  }
  float med = 0.5f * (v[0] + v[1]);           // mean of two middle order stats
  float t = (5.0f * med) / 0.6745f;
  thr_ws[0] = t;
  thr_out[0] = t;
}

// ---------------------------------------------------------------------------
// Pass 4: threshold-crossing detection for v=+x and v=-x in one sweep.
// mask = v > thr; entry i: !m[i] && m[i+1]; exit i: m[i] && !m[i+1].
// Tiles carry one halo element for the (i, i+1) pair at the tile edge.
// ---------------------------------------------------------------------------
__global__ void cross_kernel(const float* __restrict__ x, int n,
                             const float* __restrict__ thr_p,
                             unsigned* __restrict__ counts,
                             unsigned* __restrict__ lists) {
  __shared__ float buf[2][TILE + 1];
  const float thr = thr_p[0];

  const int tiles = (n + TILE - 1) / TILE;
  int t = blockIdx.x;
  int cur = 0;
  if (t < tiles && threadIdx.x == 0) {
    long long base = (long long)t * TILE;
    tdm_load_f32(buf[0], x + base, (unsigned)iminll(TILE + 1, (long long)n - base));
  }
  __syncthreads();
  for (; t < tiles; t += gridDim.x) {
    long long base = (long long)t * TILE;
    int cnt = iminll(TILE + 1, (long long)n - base);
    int nxt = t + gridDim.x;
    if (threadIdx.x == 0) {
      if (nxt < tiles) {
        long long nb = (long long)nxt * TILE;
        tdm_load_f32(buf[cur ^ 1], x + nb,
                     (unsigned)iminll(TILE + 1, (long long)n - nb));
        __builtin_amdgcn_s_wait_tensorcnt(1);
      } else {
        __builtin_amdgcn_s_wait_tensorcnt(0);
      }
    }
    __syncthreads();
    for (int i = threadIdx.x; i + 1 < cnt; i += blockDim.x) {
      float a = buf[cur][i], b = buf[cur][i + 1];
      unsigned gi = (unsigned)(base + i);       // gi <= n-2 by construction
      bool am = a > thr, bm = b > thr;
      if (!am && bm) { unsigned p = atomicAdd(&counts[0], 1u); if (p < K_MAX) lists[0u * K_MAX + p] = gi; }
      if (am && !bm) { unsigned p = atomicAdd(&counts[1], 1u); if (p < K_MAX) lists[1u * K_MAX + p] = gi; }
      bool an = (-a) > thr, bn = (-b) > thr;
      if (!an && bn) { unsigned p = atomicAdd(&counts[2], 1u); if (p < K_MAX) lists[2u * K_MAX + p] = gi; }
      if (an && !bn) { unsigned p = atomicAdd(&counts[3], 1u); if (p < K_MAX) lists[3u * K_MAX + p] = gi; }
    }
    __syncthreads();
    cur ^= 1;
  }
}

// ---------------------------------------------------------------------------
// Pairing + window-argmax + stable sort + pad. Expected counts ~7, so
// O(n^2) rank-by-count in a single block is effectively free.
// ---------------------------------------------------------------------------
__global__ void pair_kernel(const float* __restrict__ x, int n,
                            const unsigned* __restrict__ counts,
                            const unsigned* __restrict__ lists,
                            unsigned* __restrict__ sortedE,
                            unsigned* __restrict__ sortedX,
                            unsigned* __restrict__ peaks,
                            float* __restrict__ out, int entL, int extL,
                            int sign) {
  const unsigned* rawE = lists + (size_t)entL * K_MAX;
  const unsigned* rawX = lists + (size_t)extL * K_MAX;
  unsigned ne = counts[entL]; if (ne > K_MAX) ne = K_MAX;
  unsigned nx = counts[extL]; if (nx > K_MAX) nx = K_MAX;
  unsigned nn = ne < nx ? ne : nx;
  const float sgn = (float)sign;

  for (unsigned j = threadIdx.x; j < K_MAX; j += blockDim.x) out[j] = -1.0f;

  // rank-scatter sort (indices are distinct -> ranks are a permutation)
  for (unsigned j = threadIdx.x; j < ne; j += blockDim.x) {
    unsigned v = rawE[j], r = 0;
    for (unsigned m = 0; m < ne; ++m) r += (rawE[m] < v) ? 1u : 0u;
    sortedE[r] = v;
  }
  for (unsigned j = threadIdx.x; j < nx; j += blockDim.x) {
    unsigned v = rawX[j], r = 0;
    for (unsigned m = 0; m < nx; ++m) r += (rawX[m] < v) ? 1u : 0u;
    sortedX[r] = v;
  }
  __threadfence_block();
  __syncthreads();

  // pairing + first-max argmax over clipped 36-sample window
  for (unsigned j = threadIdx.x; j < K_MAX; j += blockDim.x) {
    unsigned pv = 0xFFFFFFFFu;
    if (j < nn) {
      int e = (int)sortedE[j];
      int ex = (int)sortedX[j];
      if (ex - e <= MAX_SPIKE) {
        int besto = 0;
        float best = sgn * x[e];
        for (int o = 1; o < MAX_SPIKE; ++o) {
          int idx = e + o; if (idx > n - 1) idx = n - 1;   // clipped gather
          float val = sgn * x[idx];
          if (val > best) { best = val; besto = o; }        // first max wins
        }
        pv = (unsigned)(e + besto);                         // unclipped offset
      }
    }
    peaks[j] = pv;
  }
  __threadfence_block();
  __syncthreads();

  // stable rank of kept peaks -> compacted sorted output
  for (unsigned j = threadIdx.x; j < nn; j += blockDim.x) {
    unsigned p = peaks[j];
    if (p != 0xFFFFFFFFu) {
      unsigned r = 0;
      for (unsigned m = 0; m < nn; ++m) {
        unsigned q = peaks[m];
        r += ((q < p) || (q == p && m < j)) ? 1u : 0u;
      }
      out[r] = (float)p;
    }
  }
}

__global__ void zero_kernel(unsigned* w, int cnt) {
  int i = blockIdx.x * blockDim.x + threadIdx.x;
  if (i < cnt) w[i] = 0u;
}

// ---------------------------------------------------------------------------
// workspace layout (u32 units):
//   0      hist_l1[256]
//   256    hist_l2[2][4096]
//   8448   hist_l3[2][2048]
//   12544  sel_bin1[2]   12546 sel_rank1[2]
//   12548  sel_pre2[2]   12550 sel_rank2[2]
//   12552  thr (f32)     12553 counts[4]
//   12560  lists[4][K_MAX]
//   +4K    sortedE[K_MAX], sortedX[K_MAX], peaks[K_MAX]  (shared by both signs)
// total ~1.8 MB
// ---------------------------------------------------------------------------
extern "C" void kernel_launch(void* const* d_in, const int* in_sizes, int n_in,
                              void* d_out, int out_size, void* d_ws,
                              size_t ws_size, hipStream_t stream) {
  const float* x = (const float*)d_in[0];
  const int n = in_sizes[0];
  float* out = (float*)d_out;

  unsigned* W = (unsigned*)d_ws;
  unsigned* hist1    = W;
  unsigned* hist2    = W + 256;
  unsigned* hist3    = W + 256 + 2 * 4096;
  unsigned* sel_bin1 = W + 12544;
  unsigned* sel_rk1  = W + 12546;
  unsigned* sel_pre2 = W + 12548;
  unsigned* sel_rk2  = W + 12550;
  float*    thr      = (float*)(W + 12552);
  unsigned* counts   = W + 12553;
  unsigned* lists    = W + 12560;
  unsigned* sortedE  = lists + 4u * K_MAX;
  unsigned* sortedX  = sortedE + K_MAX;
  unsigned* peaks    = sortedX + K_MAX;

  const long long k0 = (long long)n / 2 - 1;
  const long long k1 = (long long)n / 2;

  zero_kernel<<<(12560 + NTHREADS - 1) / NTHREADS, NTHREADS, 0, stream>>>(W, 12560);
  hist_l1_kernel<<<NBLOCKS, NTHREADS, 0, stream>>>(x, n, hist1);
  scan_l1_kernel<<<1, 1, 0, stream>>>(hist1, sel_bin1, sel_rk1, k0, k1);
  hist_l2_kernel<<<NBLOCKS, NTHREADS, 0, stream>>>(x, n, sel_bin1, hist2);
  scan_l2_kernel<<<1, 1, 0, stream>>>(hist2, sel_bin1, sel_rk1, sel_pre2, sel_rk2);
  hist_l3_kernel<<<NBLOCKS, NTHREADS, 0, stream>>>(x, n, sel_pre2, hist3);
  scan_l3_kernel<<<1, 1, 0, stream>>>(hist3, sel_pre2, sel_rk2, thr,
                                      out + 2 * (size_t)K_MAX);
  cross_kernel<<<NBLOCKS, NTHREADS, 0, stream>>>(x, n, thr, counts, lists);
  pair_kernel<<<1, NTHREADS, 0, stream>>>(x, n, counts, lists, sortedE, sortedX,
                                          peaks, out, 0, 1, +1);
  pair_kernel<<<1, NTHREADS, 0, stream>>>(x, n, counts, lists, sortedE, sortedX,
                                          peaks, out + K_MAX, 2, 3, -1);
}